// RelationalGATLayer_23940147708410
// MI455X (gfx1250) — compile-verified
//
#include <hip/hip_runtime.h>
#include <hip/hip_bf16.h>

// ---------------------------------------------------------------------------
// RelationalGAT on MI455X (gfx1250):
//   Phase 1: x_rel[r] = x @ W[r]  -- fp32 WMMA (V_WMMA_F32_16X16X4_F32),
//            register-blocked: one wave computes a 16x128 strip (8 col tiles),
//            A fragment loaded once per K-step and reused across 8 WMMAs.
//   Phase 2: edge attention logits + LeakyReLU + segment max (ordered atomicMax)
//   Phase 3: exp(alpha - max) + segment sum (atomicAdd)
//   Phase 4: weighted message scatter-add into out
// ---------------------------------------------------------------------------

typedef __attribute__((ext_vector_type(2))) float v2f;
typedef __attribute__((ext_vector_type(8))) float v8f;

#define DIM      128
#define NHEADS   4
#define HEADDIM  32
#define NEGSLOPE 0.2f

// ----- ordered float <-> uint mapping (monotonic, so atomicMax(uint) == float max)
__device__ __forceinline__ unsigned f2ord(float f) {
    unsigned u = __float_as_uint(f);
    return (u & 0x80000000u) ? ~u : (u | 0x80000000u);
}
__device__ __forceinline__ float ord2f(unsigned u) {
    unsigned b = (u & 0x80000000u) ? (u ^ 0x80000000u) : ~u;
    return __uint_as_float(b);
}

// ---------------------------------------------------------------------------
// Kernel 0: zero-init segment-max (ordered), denom, and output
// ---------------------------------------------------------------------------
__global__ void k_init(unsigned* __restrict__ mord, float* __restrict__ denom,
                       float* __restrict__ out, size_t nSeg, size_t nOut) {
    size_t stride = (size_t)gridDim.x * blockDim.x;
    size_t t0 = (size_t)blockIdx.x * blockDim.x + threadIdx.x;
    for (size_t i = t0; i < nSeg; i += stride) { mord[i] = 0u; denom[i] = 0.0f; }
    for (size_t i = t0; i < nOut; i += stride) out[i] = 0.0f;
}

// ---------------------------------------------------------------------------
// Kernel 1: x_rel[r] = X (NxDIM) @ W[r] (DIMxDIM), fp32 WMMA.
// One wave computes one 16x128 strip: 8 accumulators (8 x v8f = 64 VGPRs),
// K loop of 32 steps x 8 WMMAs = 256 x V_WMMA_F32_16X16X4_F32 per wave.
// A layout (16x4 f32): lane l holds A[l&15][k+koff .. k+koff+1], koff = (l>>4)*2
// B layout (4x16 f32): lane l holds B[k+koff .. +1][ct*16 + (l&15)]
// C/D layout: VGPR j, lane l -> C[j + 8*(l>=16)][l&15]
// All branches are wave-uniform => EXEC all 1s at every WMMA (ISA requirement).
// ---------------------------------------------------------------------------
__global__ __launch_bounds__(256) void k_gemm_xrel(
        const float* __restrict__ X, const float* __restrict__ W,
        float* __restrict__ xrel, int N, int R, int rowTiles) {
    const int lane = threadIdx.x & 31;
    const int wave = threadIdx.x >> 5;
    const long strip = (long)blockIdx.x * (blockDim.x >> 5) + wave; // (r, rowTile)

    const int rowTile = (int)(strip % rowTiles);
    const int r       = (int)(strip / rowTiles);
    if (r >= R) return;                                  // wave-uniform exit

    const int row0  = rowTile * 16;
    const int mLane = lane & 15;
    const int kOff  = (lane >> 4) << 1;                  // 0 or 2
    int rowA = row0 + mLane;
    if (rowA >= N) rowA = N - 1;                         // clamp (N % 16 == 0 here)

    const float* __restrict__ Arow = X + (long)rowA * DIM;
    const float* __restrict__ Wr   = W + (long)r * DIM * DIM;

    v8f acc[8];
#pragma unroll
    for (int ct = 0; ct < 8; ++ct) acc[ct] = (v8f){};

#pragma unroll 4
    for (int k = 0; k < DIM; k += 4) {
        const float2 av = *(const float2*)(Arow + k + kOff);
        v2f a; a.x = av.x; a.y = av.y;
        const float* __restrict__ Brow = Wr + (long)(k + kOff) * DIM + mLane;
#pragma unroll
        for (int ct = 0; ct < 8; ++ct) {
            v2f b;
            b.x = Brow[ct * 16];           // B[k+koff  ][ct*16 + mLane]
            b.y = Brow[DIM + ct * 16];     // B[k+koff+1][ct*16 + mLane]
            // 8 args: (neg_a, A, neg_b, B, c_mod, C, reuse_a, reuse_b)
            acc[ct] = __builtin_amdgcn_wmma_f32_16x16x4_f32(
                          false, a, false, b, (short)0, acc[ct], false, false);
        }
    }

    const int mHalf = (lane >> 4) << 3;                  // 0 or 8
    float* __restrict__ Crow = xrel + ((long)r * N + row0) * DIM;
#pragma unroll
    for (int ct = 0; ct < 8; ++ct) {
#pragma unroll
        for (int j = 0; j < 8; ++j) {
            int rr = row0 + j + mHalf;
            if (rr < N)
                Crow[(long)(j + mHalf) * DIM + ct * 16 + mLane] = acc[ct][j];
        }
    }
}

// ---------------------------------------------------------------------------
// Kernel 2: per-(edge,head) attention logit + LeakyReLU + segment max
// ---------------------------------------------------------------------------
__global__ __launch_bounds__(256) void k_edge_logits(
        const float* __restrict__ xrel, const int* __restrict__ ei,
        const int* __restrict__ et, const float* __restrict__ attn,
        float* __restrict__ alpha, unsigned* __restrict__ mord,
        int N, int E) {
    const int t = blockIdx.x * blockDim.x + threadIdx.x;
    if (t >= E * NHEADS) return;
    const int e = t >> 2, h = t & 3;
    const int src = ei[e], dst = ei[E + e], r = et[e];

    const float* __restrict__ sf = xrel + ((long)r * N + src) * DIM + h * HEADDIM;
    const float* __restrict__ df = xrel + ((long)r * N + dst) * DIM + h * HEADDIM;
    const float* __restrict__ av = attn + ((long)r * NHEADS + h) * (2 * HEADDIM);

    float acc = 0.0f;
#pragma unroll
    for (int d = 0; d < HEADDIM; d += 4) {
        float4 s  = *(const float4*)(sf + d);
        float4 a0 = *(const float4*)(av + d);
        acc += s.x * a0.x + s.y * a0.y + s.z * a0.z + s.w * a0.w;
        float4 dd = *(const float4*)(df + d);
        float4 a1 = *(const float4*)(av + HEADDIM + d);
        acc += dd.x * a1.x + dd.y * a1.y + dd.z * a1.z + dd.w * a1.w;
    }
    acc = acc > 0.0f ? acc : NEGSLOPE * acc;
    alpha[t] = acc;
    const long seg = (long)r * N + dst;
    atomicMax(&mord[seg * NHEADS + h], f2ord(acc));
}

// ---------------------------------------------------------------------------
// Kernel 3: alpha_exp = exp(alpha - segmax), segment-sum denom
// ---------------------------------------------------------------------------
__global__ __launch_bounds__(256) void k_edge_exp(
        const int* __restrict__ ei, const int* __restrict__ et,
        float* __restrict__ alpha, const unsigned* __restrict__ mord,
        float* __restrict__ denom, int N, int E) {
    const int t = blockIdx.x * blockDim.x + threadIdx.x;
    if (t >= E * NHEADS) return;
    const int e = t >> 2, h = t & 3;
    const int dst = ei[E + e], r = et[e];
    const long seg = (long)r * N + dst;
    const float mm = ord2f(mord[seg * NHEADS + h]);
    const float ex = __expf(alpha[t] - mm);
    alpha[t] = ex;                                       // reuse buffer: now alpha_exp
    atomicAdd(&denom[seg * NHEADS + h], ex);
}

// ---------------------------------------------------------------------------
// Kernel 4: out[dst] += src_feat * (alpha_exp / denom); 32 lanes per edge,
// each lane owns 4 contiguous dims (float4 gather, 4 fp32 global atomics)
// ---------------------------------------------------------------------------
__global__ __launch_bounds__(256) void k_edge_scatter(
        const float* __restrict__ xrel, const int* __restrict__ ei,
        const int* __restrict__ et, const float* __restrict__ alphaExp,
        const float* __restrict__ denom, float* __restrict__ out,
        int N, int E) {
    const long t = (long)blockIdx.x * blockDim.x + threadIdx.x;
    const long e = t >> 5;
    const int lane = (int)(t & 31);
    if (e >= E) return;
    const int src = ei[e], dst = ei[E + e], r = et[e];
    const int h = lane >> 3;                             // (lane*4)/HEADDIM
    const long seg = (long)r * N + dst;
    const float w = alphaExp[e * NHEADS + h] / denom[seg * NHEADS + h];

    const float* __restrict__ sf = xrel + ((long)r * N + src) * DIM + lane * 4;
    float4 s = *(const float4*)sf;
    float* __restrict__ o = out + (long)dst * DIM + lane * 4;
    atomicAdd(o + 0, s.x * w);
    atomicAdd(o + 1, s.y * w);
    atomicAdd(o + 2, s.z * w);
    atomicAdd(o + 3, s.w * w);
}

// ---------------------------------------------------------------------------
extern "C" void kernel_launch(void* const* d_in, const int* in_sizes, int n_in,
                              void* d_out, int out_size, void* d_ws, size_t ws_size,
                              hipStream_t stream) {
    const float* x    = (const float*)d_in[0];
    const int*   ei   = (const int*)d_in[1];   // [2, E]
    const int*   et   = (const int*)d_in[2];   // [E]
    const float* W    = (const float*)d_in[3]; // [R, DIM, DIM]
    const float* attn = (const float*)d_in[4]; // [R, H, 2*HEADDIM]
    float* out = (float*)d_out;

    const int N = in_sizes[0] / DIM;           // 100000
    const int E = in_sizes[2];                 // 600000
    const int R = in_sizes[3] / (DIM * DIM);   // 4

    // workspace layout (floats):
    //   xrel  : R*N*DIM
    //   alpha : E*NHEADS   (logits, then alpha_exp in place)
    //   mord  : R*N*NHEADS (ordered-uint segment max)
    //   denom : R*N*NHEADS
    float*    xrel  = (float*)d_ws;
    float*    alpha = xrel + (size_t)R * N * DIM;
    unsigned* mord  = (unsigned*)(alpha + (size_t)E * NHEADS);
    float*    denom = (float*)(mord + (size_t)R * N * NHEADS);

    const size_t nSeg = (size_t)R * N * NHEADS;
    const size_t nOut = (size_t)N * DIM;

    // Kernel 0: zero init
    k_init<<<2048, 256, 0, stream>>>(mord, denom, out, nSeg, nOut);

    // Kernel 1: WMMA fp32 GEMM, one wave per 16x128 strip, 8 waves per block
    const int rowTiles = (N + 15) / 16;
    const long strips = (long)R * rowTiles;
    const int gemmBlocks = (int)((strips + 7) / 8);
    k_gemm_xrel<<<gemmBlocks, 256, 0, stream>>>(x, W, xrel, N, R, rowTiles);

    // Kernel 2: logits + LeakyReLU + segment max
    const int tEH = E * NHEADS;
    k_edge_logits<<<(tEH + 255) / 256, 256, 0, stream>>>(
        xrel, ei, et, attn, alpha, mord, N, E);

    // Kernel 3: exp + segment sum
    k_edge_exp<<<(tEH + 255) / 256, 256, 0, stream>>>(
        ei, et, alpha, mord, denom, N, E);

    // Kernel 4: weighted scatter-add
    const long tScat = (long)E * 32;
    k_edge_scatter<<<(int)((tScat + 255) / 256), 256, 0, stream>>>(
        xrel, ei, et, alpha, denom, out, N, E);
}